// coupled_PINN_81372450390089
// MI455X (gfx1250) — compile-verified
//
#include <hip/hip_runtime.h>
#include <math.h>

// ---------------------------------------------------------------------------
// PINN residual (AFSD momentum + continuity) for MI455X / gfx1250.
//
// Forward-over-forward AD carried as channels through the MLPs:
//   net1 (u,v,w,p): 10 channels  (primal, d/dx_j (3), d2/dx_j dx_k (6 sym))
//   net2 (T):        4 channels  (primal, d/dx_j (3))
// Every channel multiplies the same padded 64x64 weight matrix via
// V_WMMA_F32_16X16X4_F32 (16 K-steps of K=4), 16 points per wave as the N dim.
// tanh chain rule is elementwise in the WMMA D layout (no transposes).
// Activations live in a per-wave LDS buffer, stored with even/odd feature
// interleaving so B-tile fetches are single conflict-free ds_load_b64s:
//   actIdx(row, n) = (row>>1)*32 + 2n + (row&1)
// Transcendentals use the CDNA5 hardware ops (v_tanh_f32 / v_exp_f32 /
// v_log_f32 / v_rcp_f32) instead of branchy libm paths.
// ---------------------------------------------------------------------------

typedef __attribute__((ext_vector_type(2))) float v2f;
typedef __attribute__((ext_vector_type(8))) float v8f;

#define RHO_C   2700.0f
#define EA_C    160000.0f
#define RG_C    8.314f
#define ALPHA_C 0.012f
#define LOGA_C  23.0f
#define NEXP_C  3.55f
#define LN2_C   0.69314718056f

#if __has_builtin(__builtin_amdgcn_rcpf)
#define RCPF(x) __builtin_amdgcn_rcpf(x)
#else
#define RCPF(x) (1.0f / (x))
#endif

#if __has_builtin(__builtin_amdgcn_logf)
#define LOG2F_HW(x) __builtin_amdgcn_logf(x)
#else
#define LOG2F_HW(x) (logf(x) * 1.44269504089f)
#endif

__device__ __forceinline__ float fast_tanh(float x) {
#if __has_builtin(__builtin_amdgcn_tanhf)
  return __builtin_amdgcn_tanhf(x);          // v_tanh_f32 (TRANS, co-executes)
#elif __has_builtin(__builtin_amdgcn_exp2f)
  // branch-free: tanh(x) = 1 - 2/(1 + 2^(2*log2e*x))
  const float e = __builtin_amdgcn_exp2f(x * 2.88539008178f);
  return 1.f - 2.f * RCPF(e + 1.f);
#else
  return tanhf(x);
#endif
}

struct NetP {
  const float* W[6];
  const float* b[6];
  const float* beta;
};

struct KCParams {
  const float* g;
  NetP n1, n2;
  const float* lb;
  const float* ub;
  const float* cond;   // d_ws[0]: 1.0 if mean(T) < 200 else 0.0
  float* out;
  int N;
};

// D = A(16x4) * B(4x16) + C, f32 accumulate, one wave.
__device__ __forceinline__ v8f wmma4(v2f a, v2f b, v8f c) {
  return __builtin_amdgcn_wmma_f32_16x16x4_f32(
      /*neg_a=*/false, a, /*neg_b=*/false, b,
      /*c_mod=*/(short)0, c, /*reuse_a=*/false, /*reuse_b=*/false);
}

// interleaved activation address: (row>>1)*32 + 2n + (row&1)
__device__ __forceinline__ int actIdx(int row, int n) {
  return ((row >> 1) << 5) + (n << 1) + (row & 1);
}

// z(16x16) = W[16r..16r+15][0..63] * act[0..63][0..15]   (K = 64 = 16 steps)
// A layout: lanes 0-15 hold K = 4s,4s+1 ; lanes 16-31 hold K = 4s+2,4s+3.
__device__ __forceinline__ v8f gemm64(const float* W, const float* actc,
                                      int r, int half, int n) {
  v8f acc = {};
  #pragma unroll 4
  for (int s = 0; s < 16; ++s) {
    const int k0 = 4 * s + 2 * half;
    v2f a, b;
    a.x = W[(16 * r + n) * 64 + k0];
    a.y = W[(16 * r + n) * 64 + k0 + 1];
    b = *(const v2f*)(actc + ((k0 >> 1) << 5) + (n << 1));  // one ds_load_b64
    acc = wmma4(a, b, acc);
  }
  return acc;
}

// Cooperative load of one net into zero-padded LDS (50->64 features, K pad).
__device__ void load_net_lds(float* sW0, float* sWh, float* sWL,
                             float* sB, float* sBL, float* sBeta,
                             const NetP np, int lastRows) {
  const int tid = threadIdx.x, NT = blockDim.x;
  __syncthreads();                       // previous users of the buffers done
  for (int idx = tid; idx < 64 * 4; idx += NT) {
    int r = idx >> 2, k = idx & 3;
    sW0[idx] = (r < 50 && k < 3) ? np.W[0][r * 3 + k] : 0.f;
  }
  for (int h = 0; h < 4; ++h)
    for (int idx = tid; idx < 64 * 64; idx += NT) {
      int r = idx >> 6, c = idx & 63;
      sWh[h * 4096 + idx] = (r < 50 && c < 50) ? np.W[1 + h][r * 50 + c] : 0.f;
    }
  for (int idx = tid; idx < 16 * 64; idx += NT) {
    int r = idx >> 6, c = idx & 63;
    sWL[idx] = (r < lastRows && c < 50) ? np.W[5][r * 50 + c] : 0.f;
  }
  for (int l = 0; l < 5; ++l)
    for (int idx = tid; idx < 64; idx += NT)
      sB[l * 64 + idx] = (idx < 50) ? np.b[l][idx] : 0.f;
  for (int idx = tid; idx < 16; idx += NT)
    sBL[idx] = (idx < lastRows) ? np.b[5][idx] : 0.f;
  for (int l = 0; l < 5; ++l)
    for (int idx = tid; idx < 64; idx += NT)
      sBeta[l * 64 + idx] = (idx < 50) ? np.beta[idx * 5 + l] : 0.f;   // (50,5)
  __syncthreads();
}

// ---------------------------------------------------------------------------
// Kernel C: main residual kernel.  64 threads = 2 waves; 16 points per wave.
// ---------------------------------------------------------------------------
__global__ void __launch_bounds__(64) pinn_residual_kernel(KCParams P) {
  __shared__ float sW0[64 * 4];
  __shared__ float sWh[4 * 64 * 64];
  __shared__ float sWL[16 * 64];
  __shared__ float sB[5 * 64];
  __shared__ float sBL[16];
  __shared__ float sBeta[5 * 64];
  __shared__ __align__(16) float sAct[2][10 * 64 * 16];   // [wave][ch][row/pt]

  const int lane = threadIdx.x & 31;
  const int wave = threadIdx.x >> 5;
  const int half = lane >> 4;
  const int n    = lane & 15;               // point within wave tile / WMMA col
  float* act = sAct[wave];

  const int pt  = blockIdx.x * 32 + wave * 16 + n;
  const int ptc = (pt < P.N) ? pt : (P.N - 1);
  const float X0 = P.g[3 * ptc + 0], X1 = P.g[3 * ptc + 1], X2 = P.g[3 * ptc + 2];
  const float lb0 = P.lb[0], lb1 = P.lb[1], lb2 = P.lb[2];
  const float ub0 = P.ub[0], ub1 = P.ub[1], ub2 = P.ub[2];
  const float sc0 = 2.f * RCPF(ub0 - lb0);
  const float sc1 = 2.f * RCPF(ub1 - lb1);
  const float sc2 = 2.f * RCPF(ub2 - lb2);
  const float a00 = sc0 * (X0 - lb0) - 1.f;
  const float a01 = sc1 * (X1 - lb1) - 1.f;
  const float a02 = sc2 * (X2 - lb2) - 1.f;

  // Layer-0 B tiles (K=4 x N=16) built in registers.
  // B layout: lanes 0-15 hold rows K=0,1 ; lanes 16-31 hold rows K=2,3.
  v2f Bp, Bd0, Bd1, Bd2;
  Bp.x  = half ? a02 : a00;  Bp.y  = half ? 0.f : a01;   // primal scaled coords
  Bd0.x = half ? 0.f : sc0;  Bd0.y = 0.f;                // d a0/dx row 0
  Bd1.x = 0.f;               Bd1.y = half ? 0.f : sc1;   // d a0/dy row 1
  Bd2.x = half ? sc2 : 0.f;  Bd2.y = 0.f;                // d a0/dz row 2

  constexpr int PJ[6] = {0, 0, 0, 1, 1, 2};
  constexpr int PK[6] = {0, 1, 2, 1, 2, 2};

  float u, v, w, p, Jm[4][3], S[3][6];

  // ============================ NET 1 (10 ch) ============================
  load_net_lds(sW0, sWh, sWL, sB, sBL, sBeta, P.n1, /*lastRows=*/4);
  {
    // -------- layer 0 (K = 4, single WMMA step; second-order z == 0) ------
    v8f z0[4], zj[3][4];
    #pragma unroll
    for (int r = 0; r < 4; ++r) {
      v2f aT;
      aT.x = sW0[(16 * r + n) * 4 + 2 * half];
      aT.y = sW0[(16 * r + n) * 4 + 2 * half + 1];
      v8f zero = {};
      z0[r]    = wmma4(aT, Bp,  zero);
      zj[0][r] = wmma4(aT, Bd0, zero);
      zj[1][r] = wmma4(aT, Bd1, zero);
      zj[2][r] = wmma4(aT, Bd2, zero);
    }
    float gp[4][8], gpp[4][8];
    #pragma unroll
    for (int r = 0; r < 4; ++r)
      #pragma unroll
      for (int i = 0; i < 8; ++i) {
        const int row = 16 * r + 8 * half + i;
        const float z  = z0[r][i] + sB[row];
        const float be = sBeta[row];
        const float t = fast_tanh(z), s2 = 1.f - t * t;
        act[actIdx(row, n)] = t + be * z * t;                     // channel 0
        gp[r][i]  = s2 + be * (t + z * s2);
        gpp[r][i] = s2 * (2.f * be - 2.f * t - 2.f * be * z * t);
      }
    #pragma unroll
    for (int j = 0; j < 3; ++j)
      #pragma unroll
      for (int r = 0; r < 4; ++r)
        #pragma unroll
        for (int i = 0; i < 8; ++i) {
          const int row = 16 * r + 8 * half + i;
          act[(1 + j) * 1024 + actIdx(row, n)] = gp[r][i] * zj[j][r][i];
        }
    #pragma unroll
    for (int m = 0; m < 6; ++m)
      #pragma unroll
      for (int r = 0; r < 4; ++r)
        #pragma unroll
        for (int i = 0; i < 8; ++i) {
          const int row = 16 * r + 8 * half + i;
          act[(4 + m) * 1024 + actIdx(row, n)] =
              gpp[r][i] * zj[PJ[m]][r][i] * zj[PK[m]][r][i];
        }
  }

  // -------- hidden layers 1..4 (K = 64) ---------------------------------
  for (int l = 1; l < 5; ++l) {
    const float* Wl = sWh + (l - 1) * 4096;
    v8f z0[4];
    #pragma unroll
    for (int r = 0; r < 4; ++r) z0[r] = gemm64(Wl, act, r, half, n);
    float gp[4][8], gpp[4][8];
    #pragma unroll
    for (int r = 0; r < 4; ++r)
      #pragma unroll
      for (int i = 0; i < 8; ++i) {
        const int row = 16 * r + 8 * half + i;
        const float z  = z0[r][i] + sB[l * 64 + row];
        const float be = sBeta[l * 64 + row];
        const float t = fast_tanh(z), s2 = 1.f - t * t;
        act[actIdx(row, n)] = t + be * z * t;
        gp[r][i]  = s2 + be * (t + z * s2);
        gpp[r][i] = s2 * (2.f * be - 2.f * t - 2.f * be * z * t);
      }
    v8f zj[3][4];
    #pragma unroll
    for (int j = 0; j < 3; ++j) {
      #pragma unroll
      for (int r = 0; r < 4; ++r)
        zj[j][r] = gemm64(Wl, act + (1 + j) * 1024, r, half, n);
      #pragma unroll
      for (int r = 0; r < 4; ++r)
        #pragma unroll
        for (int i = 0; i < 8; ++i) {
          const int row = 16 * r + 8 * half + i;
          act[(1 + j) * 1024 + actIdx(row, n)] = gp[r][i] * zj[j][r][i];
        }
    }
    #pragma unroll
    for (int m = 0; m < 6; ++m) {
      v8f zt[4];
      #pragma unroll
      for (int r = 0; r < 4; ++r)
        zt[r] = gemm64(Wl, act + (4 + m) * 1024, r, half, n);
      #pragma unroll
      for (int r = 0; r < 4; ++r)
        #pragma unroll
        for (int i = 0; i < 8; ++i) {
          const int row = 16 * r + 8 * half + i;
          act[(4 + m) * 1024 + actIdx(row, n)] =
              gpp[r][i] * zj[PJ[m]][r][i] * zj[PK[m]][r][i] + gp[r][i] * zt[r][i];
        }
    }
  }

  // -------- output layer (rows 0..3 = u,v,w,p; padded to 16) ------------
  {
    v8f o = gemm64(sWL, act, 0, half, n);
    u = o[0] + sBL[8 * half + 0];
    v = o[1] + sBL[8 * half + 1];
    w = o[2] + sBL[8 * half + 2];
    p = o[3] + sBL[8 * half + 3];
  }
  #pragma unroll
  for (int j = 0; j < 3; ++j) {
    v8f o = gemm64(sWL, act + (1 + j) * 1024, 0, half, n);
    Jm[0][j] = o[0]; Jm[1][j] = o[1]; Jm[2][j] = o[2]; Jm[3][j] = o[3];
  }
  #pragma unroll
  for (int m = 0; m < 6; ++m) {
    v8f o = gemm64(sWL, act + (4 + m) * 1024, 0, half, n);
    S[0][m] = o[0]; S[1][m] = o[1]; S[2][m] = o[2];
  }

  // ============================ NET 2 (4 ch) =============================
  load_net_lds(sW0, sWh, sWL, sB, sBL, sBeta, P.n2, /*lastRows=*/1);
  float T, Tg[3];
  {
    // layer 0
    v8f z0[4], zj[3][4];
    #pragma unroll
    for (int r = 0; r < 4; ++r) {
      v2f aT;
      aT.x = sW0[(16 * r + n) * 4 + 2 * half];
      aT.y = sW0[(16 * r + n) * 4 + 2 * half + 1];
      v8f zero = {};
      z0[r]    = wmma4(aT, Bp,  zero);
      zj[0][r] = wmma4(aT, Bd0, zero);
      zj[1][r] = wmma4(aT, Bd1, zero);
      zj[2][r] = wmma4(aT, Bd2, zero);
    }
    float gp2[4][8];
    #pragma unroll
    for (int r = 0; r < 4; ++r)
      #pragma unroll
      for (int i = 0; i < 8; ++i) {
        const int row = 16 * r + 8 * half + i;
        const float z  = z0[r][i] + sB[row];
        const float be = sBeta[row];
        const float t = fast_tanh(z), s2 = 1.f - t * t;
        act[actIdx(row, n)] = t + be * z * t;
        gp2[r][i] = s2 + be * (t + z * s2);
      }
    #pragma unroll
    for (int j = 0; j < 3; ++j)
      #pragma unroll
      for (int r = 0; r < 4; ++r)
        #pragma unroll
        for (int i = 0; i < 8; ++i) {
          const int row = 16 * r + 8 * half + i;
          act[(1 + j) * 1024 + actIdx(row, n)] = gp2[r][i] * zj[j][r][i];
        }
    // hidden layers
    for (int l = 1; l < 5; ++l) {
      const float* Wl = sWh + (l - 1) * 4096;
      v8f z0h[4];
      #pragma unroll
      for (int r = 0; r < 4; ++r) z0h[r] = gemm64(Wl, act, r, half, n);
      #pragma unroll
      for (int r = 0; r < 4; ++r)
        #pragma unroll
        for (int i = 0; i < 8; ++i) {
          const int row = 16 * r + 8 * half + i;
          const float z  = z0h[r][i] + sB[l * 64 + row];
          const float be = sBeta[l * 64 + row];
          const float t = fast_tanh(z), s2 = 1.f - t * t;
          act[actIdx(row, n)] = t + be * z * t;
          gp2[r][i] = s2 + be * (t + z * s2);
        }
      #pragma unroll
      for (int j = 0; j < 3; ++j) {
        v8f zt[4];
        #pragma unroll
        for (int r = 0; r < 4; ++r)
          zt[r] = gemm64(Wl, act + (1 + j) * 1024, r, half, n);
        #pragma unroll
        for (int r = 0; r < 4; ++r)
          #pragma unroll
          for (int i = 0; i < 8; ++i) {
            const int row = 16 * r + 8 * half + i;
            act[(1 + j) * 1024 + actIdx(row, n)] = gp2[r][i] * zt[r][i];
          }
      }
    }
    // output layer (row 0 = T)
    {
      v8f o = gemm64(sWL, act, 0, half, n);
      T = o[0] + sBL[8 * half];
    }
    #pragma unroll
    for (int j = 0; j < 3; ++j) {
      v8f o = gemm64(sWL, act + (1 + j) * 1024, 0, half, n);
      Tg[j] = o[0];
    }
  }

  // ============================ residual tail ============================
  const bool cond = (P.cond[0] > 0.5f);
  const float ux = Jm[0][0], uy = Jm[0][1], uz = Jm[0][2];
  const float vx = Jm[1][0], vy = Jm[1][1], vz = Jm[1][2];
  const float wx = Jm[2][0], wy = Jm[2][1], wz = Jm[2][2];
  const float px = Jm[3][0], py = Jm[3][1], pz = Jm[3][2];
  constexpr int PIDX[3][3] = {{0, 1, 2}, {1, 3, 4}, {2, 4, 5}};
  #define D2(m, j, k) S[m][PIDX[j][k]]

  const float e12s = 0.5f * (uy + vx);
  const float e13s = 0.5f * (uz + wx);
  const float e23s = 0.5f * (vz + wy);
  const float eps2 = (2.f / 3.f) * (ux * ux + vy * vy + wz * wz +
                                    2.f * (e12s * e12s + e13s * e13s + e23s * e23s));
  const float eps  = sqrtf(eps2);
  const float reps = RCPF(eps);
  const float Tb   = cond ? 300.f : T;
  const float rTb  = RCPF(Tb);
  const float logZ = LOG2F_HW(eps) * LN2_C + (EA_C / RG_C) * rTb;
  const float W_   = (logZ - LOGA_C) * (1.f / NEXP_C);
  const float sigma = (LN2_C / NEXP_C + W_) * (1.f / ALPHA_C);
  const float mu    = sigma * (1.f / 3.f) * reps;

  float dmu[3];
  #pragma unroll
  for (int k = 0; k < 3; ++k) {
    const float dss = (4.f / 3.f) * (ux * D2(0, 0, k) + vy * D2(1, 1, k) + wz * D2(2, 2, k)
                    + e12s * (D2(0, 1, k) + D2(1, 0, k))
                    + e13s * (D2(0, 2, k) + D2(2, 0, k))
                    + e23s * (D2(1, 2, k) + D2(2, 1, k)));
    const float deps = 0.5f * dss * reps;
    const float dT   = cond ? 0.f : Tg[k];
    const float dsig = (1.f / (ALPHA_C * NEXP_C)) *
                       (deps * reps - (EA_C / RG_C) * rTb * rTb * dT);
    dmu[k] = (dsig * eps - sigma * deps) * (1.f / 3.f) * reps * reps;
  }
  #define HH(m9, k) (dmu[k] * Jm[(m9) / 3][(m9) % 3] + mu * S[(m9) / 3][PIDX[(m9) % 3][k]])

  const float f1 = RHO_C * 1e-6f * (2.f * u * ux + (v * uy + u * vy) + (w * uz + u * wz)) + px
                 - 2.f * HH(0, 0) - (HH(3, 1) + HH(1, 1)) - (HH(6, 2) + HH(2, 2));
  const float f2 = RHO_C * 1e-6f * ((v * ux + u * vx) + 2.f * v * vy + (w * vz + v * wz)) + py
                 - (HH(1, 0) + HH(3, 0)) - 2.f * HH(4, 1) - (HH(7, 2) + HH(5, 2));
  const float f3 = RHO_C * 1e-6f * ((w * ux + u * wx) + (w * vy + v * wy) + 2.f * w * wz) + pz
                 - (HH(2, 0) + HH(6, 0)) - (HH(5, 1) + HH(7, 1)) - 2.f * HH(8, 2);
  const float f4 = ux + vy + wz;

  if (half == 0 && pt < P.N) {
    float* o = P.out + 4 * pt;
    o[0] = f1; o[1] = f2; o[2] = f3; o[3] = f4;
  }
  #undef HH
  #undef D2
}

// ---------------------------------------------------------------------------
// Kernel A: per-block partial sums of T(x) (primal net2 only, plain VALU;
// <7% of total FLOPs).  Deterministic intra-block tree reduction.
// ---------------------------------------------------------------------------
__global__ void __launch_bounds__(256) tmean_kernel(const float* g, NetP n2,
                                                    const float* lb, const float* ub,
                                                    float* partial, int N) {
  __shared__ float sW0[150];
  __shared__ float sWh[4][2500];
  __shared__ float sWL[50];
  __shared__ float sb[5][50];
  __shared__ float sbe[5][50];
  __shared__ float sbL;
  __shared__ float red[256];
  const int tid = threadIdx.x;
  for (int i = tid; i < 150; i += 256) sW0[i] = n2.W[0][i];
  for (int h = 0; h < 4; ++h)
    for (int i = tid; i < 2500; i += 256) sWh[h][i] = n2.W[1 + h][i];
  for (int i = tid; i < 50; i += 256) sWL[i] = n2.W[5][i];
  for (int l = 0; l < 5; ++l)
    for (int i = tid; i < 50; i += 256) {
      sb[l][i]  = n2.b[l][i];
      sbe[l][i] = n2.beta[i * 5 + l];
    }
  if (tid == 0) sbL = n2.b[5][0];
  __syncthreads();

  const float lb0 = lb[0], lb1 = lb[1], lb2 = lb[2];
  const float s0 = 2.f * RCPF(ub[0] - lb0);
  const float s1 = 2.f * RCPF(ub[1] - lb1);
  const float s2 = 2.f * RCPF(ub[2] - lb2);
  float sum = 0.f;
  for (int pt = blockIdx.x * 256 + tid; pt < N; pt += gridDim.x * 256) {
    const float x0 = s0 * (g[3 * pt + 0] - lb0) - 1.f;
    const float x1 = s1 * (g[3 * pt + 1] - lb1) - 1.f;
    const float x2 = s2 * (g[3 * pt + 2] - lb2) - 1.f;
    float a[50], z[50];
    for (int f = 0; f < 50; ++f) {
      const float zz = sW0[f * 3] * x0 + sW0[f * 3 + 1] * x1 + sW0[f * 3 + 2] * x2 + sb[0][f];
      const float t = fast_tanh(zz);
      a[f] = t + sbe[0][f] * zz * t;
    }
    for (int l = 1; l < 5; ++l) {
      for (int f = 0; f < 50; ++f) {
        float zz = sb[l][f];
        for (int k = 0; k < 50; ++k) zz += sWh[l - 1][f * 50 + k] * a[k];
        z[f] = zz;
      }
      for (int f = 0; f < 50; ++f) {
        const float t = fast_tanh(z[f]);
        a[f] = t + sbe[l][f] * z[f] * t;
      }
    }
    float T = sbL;
    for (int k = 0; k < 50; ++k) T += sWL[k] * a[k];
    sum += T;
  }
  red[tid] = sum;
  __syncthreads();
  for (int s = 128; s > 0; s >>= 1) {
    if (tid < s) red[tid] += red[tid + s];
    __syncthreads();
  }
  if (tid == 0) partial[blockIdx.x] = red[0];
}

// Kernel B: ordered final reduction -> cond flag (deterministic).
__global__ void cond_kernel(const float* partial, float* condOut, int nblk, int N) {
  if (blockIdx.x == 0 && threadIdx.x == 0) {
    float s = 0.f;
    for (int i = 0; i < nblk; ++i) s += partial[i];
    condOut[0] = ((s / (float)N) < 200.f) ? 1.f : 0.f;
  }
}

// ---------------------------------------------------------------------------
// d_in layout (flattened setup_inputs dict):
//   0: g | 1-6: Ws1[0..5] | 7-12: bs1[0..5] | 13: beta1
//   14-19: Ws2[0..5] | 20-25: bs2[0..5] | 26: beta2 | 27: lb | 28: ub
// ---------------------------------------------------------------------------
extern "C" void kernel_launch(void* const* d_in, const int* in_sizes, int n_in,
                              void* d_out, int out_size, void* d_ws, size_t ws_size,
                              hipStream_t stream) {
  (void)n_in; (void)out_size; (void)ws_size;
  KCParams P;
  P.g = (const float*)d_in[0];
  for (int i = 0; i < 6; ++i) {
    P.n1.W[i] = (const float*)d_in[1 + i];
    P.n1.b[i] = (const float*)d_in[7 + i];
    P.n2.W[i] = (const float*)d_in[14 + i];
    P.n2.b[i] = (const float*)d_in[20 + i];
  }
  P.n1.beta = (const float*)d_in[13];
  P.n2.beta = (const float*)d_in[26];
  P.lb = (const float*)d_in[27];
  P.ub = (const float*)d_in[28];
  const int N = in_sizes[0] / 3;
  P.N = N;
  float* ws = (float*)d_ws;
  P.cond = ws;                 // ws[0] = cond flag
  P.out = (float*)d_out;

  const int NBLKA = 128;
  tmean_kernel<<<NBLKA, 256, 0, stream>>>(P.g, P.n2, P.lb, P.ub, ws + 1, N);
  cond_kernel<<<1, 32, 0, stream>>>(ws + 1, ws, NBLKA, N);
  const int nblocks = (N + 31) / 32;     // 32 points per block (2 waves x 16)
  pinn_residual_kernel<<<nblocks, 64, 0, stream>>>(P);
}